// E45PureAccumulationCell_47656957116862
// MI455X (gfx1250) — compile-verified
//
#include <hip/hip_runtime.h>
#include <hip/hip_bf16.h>

// Problem constants (from reference): T=4096, B=8, D=1024
#define T_LEN 4096
#define BD    8192          // B*D
#define CC    64            // number of time chunks
#define LL    64            // chunk length (T_LEN / CC), multiple of 16

typedef __attribute__((ext_vector_type(2))) float v2f;
typedef __attribute__((ext_vector_type(8))) float v8f;

// Fast sigmoid: single v_rcp_f32 instead of IEEE div sequence (~1 ULP, fine here).
__device__ __forceinline__ float dev_sigmoid(float v) {
    return __builtin_amdgcn_rcpf(1.0f + __expf(-v));
}

// alpha^e by repeated multiply (e <= 64); runs once per thread at kernel start.
__device__ __forceinline__ float apow(float a, int e) {
    float p = 1.0f;
    for (int i = 0; i < e; ++i) p *= a;
    return p;
}

// ---------------------------------------------------------------------------
// Kernel 1: per-chunk decayed sum  s_c[n] = sum_{i<L} alpha^(L-1-i) x[cL+i, n]
// (== chunk-end h assuming h_in = 0).  Each thread owns 4 channels -> b128
// loads, coalesced across lanes, unrolled so loads batch ahead of the FMA chain.
// ---------------------------------------------------------------------------
__global__ void k_chunk_sum(const float4* __restrict__ x4,
                            const float* __restrict__ la,
                            float4* __restrict__ s4) {
    const float alpha = dev_sigmoid(la[0]);
    const int n4 = blockIdx.x * blockDim.x + threadIdx.x;   // channel/4
    const int c  = blockIdx.y;                              // chunk
    const float4* xp = x4 + (size_t)c * LL * (BD / 4) + n4;
    float4 acc = {0.0f, 0.0f, 0.0f, 0.0f};
#pragma unroll 8
    for (int i = 0; i < LL; ++i) {
        const float4 v = xp[(size_t)i * (BD / 4)];
        acc.x = fmaf(alpha, acc.x, v.x);
        acc.y = fmaf(alpha, acc.y, v.y);
        acc.z = fmaf(alpha, acc.z, v.z);
        acc.w = fmaf(alpha, acc.w, v.w);
    }
    s4[(size_t)c * (BD / 4) + n4] = acc;
}

// ---------------------------------------------------------------------------
// Kernel 2: serial scan across the CC chunks (cheap: 64 steps, 2048 threads).
// carries[c][n] = h entering chunk c.  Also writes h_out[0] = h0.
// ---------------------------------------------------------------------------
__global__ void k_chunk_scan(const float4* __restrict__ s4,
                             const float4* __restrict__ h04,
                             const float* __restrict__ la,
                             float4* __restrict__ carries4,
                             float4* __restrict__ hout4) {
    const float alpha = dev_sigmoid(la[0]);
    const float aL = apow(alpha, LL);
    const int n4 = blockIdx.x * blockDim.x + threadIdx.x;
    float4 carry = h04[n4];
    hout4[n4] = carry;                     // h[0] row
    for (int c = 0; c < CC; ++c) {
        carries4[(size_t)c * (BD / 4) + n4] = carry;
        const float4 v = s4[(size_t)c * (BD / 4) + n4];
        carry.x = fmaf(aL, carry.x, v.x);
        carry.y = fmaf(aL, carry.y, v.y);
        carry.z = fmaf(aL, carry.z, v.z);
        carry.w = fmaf(aL, carry.w, v.w);
    }
}

// ---------------------------------------------------------------------------
// Kernel 3: main pass.  Each wave owns 16 channels and one chunk of 64 steps,
// processed as 4 time-tiles of 16 via V_WMMA_F32_16X16X4_F32:
//     H(16x16) = A_scan(16x16) * X(16x16) + C,   A_scan[i][k] = alpha^(i-k) (k<=i)
//     C[i][n]  = alpha^(i+1) * h_in[n]
// A operand (documented layout): lane&15 = M; VGPR0 holds K = 4q + (0|2),
// VGPR1 holds K = 4q + (1|3) split by lane half.
// B operand (X tile): lane&15 = N (channel), same K split as A.
// D (C/D documented layout): VGPR r -> time row r (lanes 0-15) / r+8 (16-31),
// lane&15 = channel  -> coalesced 64B stores per half-wave.
// Tile carry = D row M=15 (VGPR7, lanes 16-31), broadcast via wave32 shuffle.
// ---------------------------------------------------------------------------
__global__ void k_wmma_scan(const float* __restrict__ x,
                            const float* __restrict__ carries,
                            const float* __restrict__ la,
                            float* __restrict__ out,
                            float* __restrict__ hout) {
    const float alpha = dev_sigmoid(la[0]);

    const int lane  = threadIdx.x & 31;
    const int wave  = threadIdx.x >> 5;            // 0..7
    const int half  = lane >> 4;                   // 0 | 1
    const int mrow  = lane & 15;                   // M for A, N for B/D
    const int chunk = blockIdx.x;                  // 0..CC-1
    const int tileN = blockIdx.y * 8 + wave;       // 0..BD/16-1
    const int n0    = tileN * 16;
    const int n     = n0 + mrow;

    // Constant A-matrix slices (scan operator), per documented A layout.
    float A0[4], A1[4];
#pragma unroll
    for (int q = 0; q < 4; ++q) {
        const int k0 = 4 * q + 2 * half;
        const int k1 = k0 + 1;
        A0[q] = (k0 <= mrow) ? apow(alpha, mrow - k0) : 0.0f;
        A1[q] = (k1 <= mrow) ? apow(alpha, mrow - k1) : 0.0f;
    }
    // Carry-injection coefficients alpha^(m+1) for the C matrix rows.
    float Cc[8];
#pragma unroll
    for (int r = 0; r < 8; ++r) Cc[r] = apow(alpha, r + 8 * half + 1);

    float carry = carries[(size_t)chunk * BD + n];

    for (int tile = 0; tile < LL / 16; ++tile) {
        const int t0 = chunk * LL + tile * 16;

        // Prefetch next tile's x rows (in-bounds: stays inside this chunk).
        if (tile + 1 < LL / 16) {
            __builtin_prefetch(x + (size_t)(t0 + 16 + 2 * half) * BD + n, 0, 3);
        }

        v8f acc;
#pragma unroll
        for (int r = 0; r < 8; ++r) acc[r] = Cc[r] * carry;

#pragma unroll
        for (int q = 0; q < 4; ++q) {
            const int kb0 = 4 * q + 2 * half;
            v2f a; a.x = A0[q]; a.y = A1[q];
            v2f b;
            b.x = x[(size_t)(t0 + kb0)     * BD + n];
            b.y = x[(size_t)(t0 + kb0 + 1) * BD + n];
            acc = __builtin_amdgcn_wmma_f32_16x16x4_f32(
                /*neg_a=*/false, a, /*neg_b=*/false, b,
                /*c_mod=*/(short)0, acc, /*reuse_a=*/false, /*reuse_b=*/false);
        }

        // New carry = H row M=15 -> VGPR 7 of lanes 16..31; broadcast to both halves.
        carry = __shfl(acc[7], 16 + mrow, 32);

#pragma unroll
        for (int r = 0; r < 8; ++r) {
            const int t = t0 + r + 8 * half;
            const float hv = acc[r];
            const float sg = dev_sigmoid(hv);              // v_exp + v_rcp
            out[(size_t)t * BD + n]         = hv * hv * sg; // hs * silu(hs)
            hout[(size_t)(t + 1) * BD + n]  = hv;
        }
    }
}

// ---------------------------------------------------------------------------
extern "C" void kernel_launch(void* const* d_in, const int* in_sizes, int n_in,
                              void* d_out, int out_size, void* d_ws, size_t ws_size,
                              hipStream_t stream) {
    const float* x  = (const float*)d_in[0];   // [T, B, D]
    const float* h0 = (const float*)d_in[1];   // [B, D]
    const float* la = (const float*)d_in[2];   // scalar log_alpha

    float* out  = (float*)d_out;               // [T, B, D]
    float* hout = out + (size_t)T_LEN * BD;    // [T+1, B, D]

    float* s       = (float*)d_ws;             // [CC, BD]
    float* carries = s + (size_t)CC * BD;      // [CC, BD]

    k_chunk_sum <<<dim3(BD / 4 / 256, CC), 256, 0, stream>>>(
        (const float4*)x, la, (float4*)s);
    k_chunk_scan<<<dim3(BD / 4 / 256), 256, 0, stream>>>(
        (const float4*)s, (const float4*)h0, la, (float4*)carries, (float4*)hout);
    k_wmma_scan <<<dim3(CC, BD / 128), 256, 0, stream>>>(
        x, carries, la, out, hout);
}